// SimplePolyModel_43293270344154
// MI455X (gfx1250) — compile-verified
//
#include <hip/hip_runtime.h>
#include <hip/hip_bf16.h>

// r = relu( Σ_m q[m]·s[m] + b ) per row, where
//   s[m] = Σ_l x[l]·w[m+l]   (GEMM vs shared Hankel(w)  -> f32 WMMA, register-blocked)
//   q[m] = Σ_{i+j=m} x[i]x[j] (per-row self-conv, symmetry-halved -> VALU)
//
// One block = 16 rows, 256 threads (8 wave32). Fully fused in LDS.

typedef float v2f __attribute__((ext_vector_type(2)));
typedef float v8f __attribute__((ext_vector_type(8)));

#define NN 1024          // coeffs per poly
#define XS_PITCH 1028    // x tile row pitch (floats): 4-bank row stride -> conflict-free, 16B aligned
#define Q_PITCH 2064     // q row pitch: wave halves hit disjoint bank groups
#define WL_LEN 3080      // w in LDS, zero-padded past 3069 (max index used = 3070)
#define MBLK 16          // m-blocks per wave (register-blocked accumulators)

__global__ __launch_bounds__(256)
void poly_cube_dot_kernel(const float* __restrict__ x,
                          const float* __restrict__ w,
                          const float* __restrict__ bias,
                          float* __restrict__ out)
{
    extern __shared__ float smem[];
    float* xs     = smem;                       // 16 * XS_PITCH
    float* wl     = xs + 16 * XS_PITCH;         // WL_LEN
    float* q      = wl + WL_LEN;                // 16 * Q_PITCH
    float* rowacc = q + 16 * Q_PITCH;           // 16

    const int tid  = threadIdx.x;
    const int lane = tid & 31;
    const int wave = tid >> 5;
    const int half = lane >> 4;   // 0: lanes 0-15, 1: lanes 16-31
    const int lm   = lane & 15;

    const long long rowBase = (long long)blockIdx.x * 16;

    // gfx1250 prefetch of this tile's x rows (global_prefetch_b8 path)
    if (tid < 16) __builtin_prefetch(&x[(rowBase + tid) * NN], 0, 3);

    // ---- stage x tile into LDS with b128 transfers (coalesced) ----
    for (int e = tid; e < 16 * (NN / 4); e += 256) {
        const int r = e >> 8, c4 = e & 255;
        const float4 v = ((const float4*)x)[(rowBase + r) * (NN / 4) + c4];
        *(float4*)&xs[r * XS_PITCH + 4 * c4] = v;
    }
    // ---- stage w into LDS, zero-padded ----
    for (int e = tid; e < WL_LEN; e += 256)
        wl[e] = (e < 3070) ? w[e] : 0.0f;
    if (tid < 16) rowacc[tid] = 0.0f;
    __syncthreads();

    // ---- per-row self-conv q[m], symmetry-halved (VALU) ----
    for (int e = tid; e < 16 * 2048; e += 256) {
        const int r = e >> 11;
        const int m = e & 2047;
        float acc = 0.0f;
        if (m <= 2046) {
            const int lo  = m > (NN - 1) ? m - (NN - 1) : 0;
            const int mid = (m + 1) >> 1;          // pairs i < m-i
            const float* xr = &xs[r * XS_PITCH];
            float s = 0.0f;
            for (int i = lo; i < mid; ++i)
                s = __builtin_fmaf(xr[i], xr[m - i], s);
            acc = s + s;
            if (!(m & 1)) {
                const float c = xr[m >> 1];
                acc = __builtin_fmaf(c, c, acc);
            }
        }
        q[r * Q_PITCH + m] = acc;   // q[r][2047] = 0 (kills garbage s[2047])
    }
    __syncthreads();

    // ---- WMMA phase: S tiles = X(16xK) * Hankel(w)(Kx16), K = 1024 ----
    // Register-blocked: one wave holds 16 accumulator tiles (m-blocks); the A
    // fragment (x data) is loaded once per K-step and reused across all 16.
    // Per K-step: phase 1 loads all 16 B fragments into distinct regs (keeps
    // many ds_loads in flight, staged s_wait_dscnt), phase 2 issues 16 WMMAs.
    // A 16x4 f32 layout: lanes 0-15 = M rows; vgpr0 K = 2*half, vgpr1 K = 2*half+1.
    // B 4x16 f32 mirrors: lane%16 = N, same K striping per lane-half.
    v8f acc[MBLK];
#pragma unroll
    for (int t = 0; t < MBLK; ++t) acc[t] = (v8f){};

    const float* xrowA = &xs[lm * XS_PITCH];
    const int kofs  = 2 * half;
    const int wbase = wave * (MBLK * 16) + lm + kofs;   // + 16*t + l0 -> B element

    for (int l0 = 0; l0 < NN; l0 += 4) {
        const float* wp = &wl[wbase + l0];
        v2f bm[MBLK];
#pragma unroll
        for (int t = 0; t < MBLK; ++t) {          // phase 1: batch LDS loads
            bm[t].x = wp[16 * t];
            bm[t].y = wp[16 * t + 1];
        }
        v2f a;  a.x = xrowA[l0 + kofs];  a.y = xrowA[l0 + kofs + 1];
#pragma unroll
        for (int t = 0; t < MBLK; ++t) {          // phase 2: matrix pipe
            acc[t] = __builtin_amdgcn_wmma_f32_16x16x4_f32(
                         /*neg_a=*/false, a, /*neg_b=*/false, bm[t],
                         /*c_mod=*/(short)0, acc[t], /*reuse_a=*/false, /*reuse_b=*/false);
        }
    }

    // ---- dot accumulated s tiles against q, reduce per row ----
    // acc[t][v] = s[row = v + 8*half][16*(wave*16+t) + lm]
    float partial[8];
#pragma unroll
    for (int v = 0; v < 8; ++v) partial[v] = 0.0f;

#pragma unroll
    for (int t = 0; t < MBLK; ++t) {
        const int m0 = (wave * MBLK + t) * 16;
#pragma unroll
        for (int v = 0; v < 8; ++v) {
            const int row = v + 8 * half;
            partial[v] = __builtin_fmaf(acc[t][v], q[row * Q_PITCH + m0 + lm], partial[v]);
        }
    }

    // ---- per-row reduction via LDS float atomics (ds_add_f32) ----
#pragma unroll
    for (int v = 0; v < 8; ++v)
        atomicAdd(&rowacc[v + 8 * half], partial[v]);
    __syncthreads();

    if (tid < 16) {
        float r = rowacc[tid] + bias[0];
        out[rowBase + tid] = r > 0.0f ? r : 0.0f;
    }
}

extern "C" void kernel_launch(void* const* d_in, const int* in_sizes, int n_in,
                              void* d_out, int out_size, void* d_ws, size_t ws_size,
                              hipStream_t stream) {
    const float* x    = (const float*)d_in[0];   // [32768, 1024]
    const float* w    = (const float*)d_in[1];   // [3070]
    const float* bias = (const float*)d_in[2];   // [1]
    float* out = (float*)d_out;                  // [32768]

    const int B = in_sizes[0] / NN;              // 32768
    const int grid = B / 16;                     // 2048 blocks

    const size_t lds_bytes =
        (size_t)(16 * XS_PITCH + WL_LEN + 16 * Q_PITCH + 16) * sizeof(float); // ~210 KB of 320 KB/WGP

    poly_cube_dot_kernel<<<grid, 256, lds_bytes, stream>>>(x, w, bias, out);
}